// Ewald_34419867910906
// MI455X (gfx1250) — compile-verified
//
#include <hip/hip_runtime.h>
#include <math.h>

// ---------------------------------------------------------------------------
// Ewald real-space potential, n = 4096 atoms, single scalar output.
//   pot = 0.5 * sum_{i!=j} q_i q_j * erf(d_ij/sqrt(2)) / d_ij * (90.0474/2pi)
//
// d^2_ij = |r_i|^2 + |r_j|^2 - 2 r_i.r_j  is computed one 16x16 tile per
// V_WMMA_F32_16X16X4_F32 (K=4 holds x,y,z + constant lane), so the matrix
// pipe produces the squared-distance tile and the VALU runs a fully
// branchless rsq + rcp + exp erf evaluation per pair.
// ---------------------------------------------------------------------------

typedef float v2f __attribute__((ext_vector_type(2)));
typedef float v8f __attribute__((ext_vector_type(8)));

#define N_ATOMS          4096
#define TILE             16
#define NT               (N_ATOMS / TILE)        // 256 tiles per dimension
#define WAVES_PER_BLOCK  8
#define NBLOCKS          512
#define TOTAL_WAVES      (NBLOCKS * WAVES_PER_BLOCK)   // 4096 waves
#define TILES_PER_WAVE   ((NT * NT) / TOTAL_WAVES)     // 16 tiles / wave

// Branchless erf for x >= 0 (Abramowitz & Stegun 7.1.26, |err| < 1.5e-7):
//   erf(x) = 1 - (a1 t + a2 t^2 + a3 t^3 + a4 t^4 + a5 t^5) exp(-x^2),
//   t = 1 / (1 + p x)
// rcp -> v_rcp_f32, exp -> v_exp_f32: two hardware transcendentals, no branches.
__device__ __forceinline__ float erf_pos(float x) {
    const float p  = 0.3275911f;
    const float a1 = 0.254829592f;
    const float a2 = -0.284496736f;
    const float a3 = 1.421413741f;
    const float a4 = -1.453152027f;
    const float a5 = 1.061405429f;
    float t    = __builtin_amdgcn_rcpf(fmaf(p, x, 1.0f));
    float poly = fmaf(fmaf(fmaf(fmaf(a5, t, a4), t, a3), t, a2), t, a1) * t;
    float ex   = __expf(-x * x);                 // v_exp_f32
    return fmaf(-poly, ex, 1.0f);
}

__global__ __launch_bounds__(WAVES_PER_BLOCK * 32)
void ewald_tiles_kernel(const float* __restrict__ q,
                        const float* __restrict__ r,
                        float* __restrict__ partial)
{
    const float A_ERF = 0.70710678118654752f;    // 1/(sigma*sqrt(2))
    const float NORMC = 90.0474f / (2.0f * 3.14159265358979323846f);

    const int lane  = threadIdx.x & 31;
    const int wave  = threadIdx.x >> 5;
    const int gwave = blockIdx.x * WAVES_PER_BLOCK + wave;

    // Each wave owns one fixed 16-row block (ti) and 16 consecutive column tiles.
    const int ti  = gwave >> 4;                    // 0..255
    const int tj0 = (gwave & 15) * TILES_PER_WAVE; // 0,16,...,240
    const int I   = ti * TILE;

    const bool hi   = lane >= 16;                  // upper half-wave: K=2,3 / rows M+8
    const int  mrow = lane & 15;

    // ---- Row-block data (loaded once per wave) ----
    const int irow = I + mrow;
    const float rxi = r[3 * irow + 0];
    const float ryi = r[3 * irow + 1];
    const float rzi = r[3 * irow + 2];
    const float n2row = fmaf(rxi, rxi, fmaf(ryi, ryi, rzi * rzi));
    const float qrow  = q[irow];

    // A operand (16x4 f32): lanes 0-15 hold K=0,1 ; lanes 16-31 hold K=2,3.
    // Row i of A = (-2x_i, -2y_i, -2z_i, 1). Branchless selects -> v_cndmask.
    v2f Aop;
    Aop.x = hi ? (-2.0f * rzi) : (-2.0f * rxi);
    Aop.y = hi ? 1.0f          : (-2.0f * ryi);

    // Per-accumulator-register row broadcasts: VGPR k holds row I+k (lanes 0-15)
    // or row I+k+8 (lanes 16-31).
    float n2i[8], qi[8];
#pragma unroll
    for (int k = 0; k < 8; ++k) {
        const int src = k + (hi ? 8 : 0);          // lane src holds row I+src data
        n2i[k] = __shfl(n2row, src, 32);
        qi[k]  = __shfl(qrow,  src, 32);
    }

    float acc = 0.0f;

    for (int t = 0; t < TILES_PER_WAVE; ++t) {
        const int J    = (tj0 + t) * TILE;
        const int jcol = J + mrow;

        // Column-block data (all lanes compute everything; selects are branchless).
        // B operand (4x16 f32): lanes 0-15 hold K=0,1 of column n = lane;
        // lanes 16-31 hold K=2,3 of column n = lane-16.
        // Col j of B = (x_j, y_j, z_j, |r_j|^2).
        const float bx  = r[3 * jcol + 0];
        const float by  = r[3 * jcol + 1];
        const float bz  = r[3 * jcol + 2];
        const float n2j = fmaf(bx, bx, fmaf(by, by, bz * bz));
        const float qj  = q[jcol];

        v2f Bop;
        Bop.x = hi ? bz  : bx;
        Bop.y = hi ? n2j : by;

        // C initialized with |r_i|^2 per row -> D = d^2 tile.
        v8f C;
#pragma unroll
        for (int k = 0; k < 8; ++k) C[k] = n2i[k];

        v8f D = __builtin_amdgcn_wmma_f32_16x16x4_f32(
            /*neg_a=*/false, Aop, /*neg_b=*/false, Bop,
            /*c_mod=*/(short)0, C, /*reuse_a=*/false, /*reuse_b=*/false);

        // Element-wise, fully branchless:
        //   erf(d*a)/d = erf_pos(d*a) * rsqrt(d^2), masked by (i != j).
#pragma unroll
        for (int k = 0; k < 8; ++k) {
            const int i = I + k + (hi ? 8 : 0);
            const int j = J + mrow;
            const float mask = (i != j) ? 1.0f : 0.0f;   // v_cndmask
            float d2   = fmaxf(D[k], 1e-12f);
            float rinv = rsqrtf(d2);                     // v_rsq_f32
            float dist = d2 * rinv;
            float e    = erf_pos(dist * A_ERF);
            acc = fmaf(mask * qi[k] * qj, e * rinv, acc);
        }
    }

    acc *= 0.5f * NORMC;

    // ---- wave32 reduction ----
#pragma unroll
    for (int off = 16; off > 0; off >>= 1)
        acc += __shfl_xor(acc, off, 32);

    __shared__ float wsum[WAVES_PER_BLOCK];
    if (lane == 0) wsum[wave] = acc;
    __syncthreads();

    if (threadIdx.x == 0) {
        float s = 0.0f;
#pragma unroll
        for (int w = 0; w < WAVES_PER_BLOCK; ++w) s += wsum[w];
        partial[blockIdx.x] = s;
    }
}

// Deterministic final reduction of the 512 per-block partials.
__global__ __launch_bounds__(256)
void ewald_reduce_kernel(const float* __restrict__ partial,
                         float* __restrict__ out)
{
    __shared__ float s[256];
    const int t = threadIdx.x;
    s[t] = partial[t] + partial[t + 256];
    __syncthreads();
#pragma unroll
    for (int off = 128; off > 0; off >>= 1) {
        if (t < off) s[t] += s[t + off];
        __syncthreads();
    }
    if (t == 0) out[0] = s[0];
}

extern "C" void kernel_launch(void* const* d_in, const int* in_sizes, int n_in,
                              void* d_out, int out_size, void* d_ws, size_t ws_size,
                              hipStream_t stream) {
    // setup_inputs order: q [4096] f32, r [4096*3] f32, cell [9] f32, batch [4096] i32
    const float* q = (const float*)d_in[0];
    const float* r = (const float*)d_in[1];
    float* out     = (float*)d_out;
    float* partial = (float*)d_ws;               // 512 floats = 2 KB scratch

    ewald_tiles_kernel<<<NBLOCKS, WAVES_PER_BLOCK * 32, 0, stream>>>(q, r, partial);
    ewald_reduce_kernel<<<1, 256, 0, stream>>>(partial, out);
}